// ROIEmbedding1D_69295002353911
// MI455X (gfx1250) — compile-verified
//
#include <hip/hip_runtime.h>

// ROIEmbedding1D multi-resolution max pool: x[4096,16384] f32 -> out[4096,31] f32.
// Memory-bound streaming reduction (256 MB read -> ~11 us floor at 23.3 TB/s).
// CDNA5 path: async HBM->LDS copy engine (global_load_async_to_lds_b128 +
// s_wait_asynccnt), triple-buffered with issue distance 2, one barrier/tile.

#define WIDTH    16384
#define TILE     2048              // floats per tile (8 KB)
#define NTILES   (WIDTH / TILE)    // 8
#define NBUF     3                 // triple buffer; issue distance 2 (K >= D+1)
#define NTHREADS 256               // 8 wave32s
#define NSEG     16                // finest pooling: 16 segments of 1024 floats
#define NOUT     31                // 1+2+4+8+16

__global__ __launch_bounds__(NTHREADS)
void roi_embed_max_kernel(const float* __restrict__ x,
                          float* __restrict__ out,
                          int batch)
{
    __shared__ __attribute__((aligned(16))) float buf[NBUF][TILE]; // 24 KB
    __shared__ float waveres[NSEG * 8];  // [segment][wave-chunk] partial maxes
    __shared__ float outbuf[NOUT];

    const int row  = blockIdx.x;
    if (row >= batch) return;            // whole block exits together (before barriers)

    const int tid  = threadIdx.x;
    const int wave = tid >> 5;
    const int lane = tid & 31;

    const float* rowp = x + (size_t)row * WIDTH;

    // LDS byte offsets of this thread's float4 slot in each buffer.
    // Low 32 bits of a generic shared pointer == LDS byte offset (ISA aperture rule).
    unsigned ldsb[NBUF];
    #pragma unroll
    for (int k = 0; k < NBUF; ++k)
        ldsb[k] = (unsigned)(size_t)(&buf[k][0]) + (unsigned)tid * 16u;

    // Issue two fully-coalesced async b128 loads for tile t into buf[t % NBUF].
    // Issue 0 covers tile floats [0,1024)    -> segment 2t
    // Issue 1 covers tile floats [1024,2048) -> segment 2t+1
    auto issue_tile = [&](int t) {
        const unsigned lbase = ldsb[t % NBUF];
        unsigned long long g0 =
            (unsigned long long)(const void*)(rowp + (size_t)t * TILE + tid * 4);
        unsigned long long g1 = g0 + (TILE / 2) * sizeof(float);
        unsigned l0 = lbase;
        unsigned l1 = lbase + (TILE / 2) * sizeof(float);
        asm volatile("global_load_async_to_lds_b128 %0, %1, off"
                     :: "v"(l0), "v"(g0) : "memory");
        asm volatile("global_load_async_to_lds_b128 %0, %1, off"
                     :: "v"(l1), "v"(g1) : "memory");
    };

    // Prologue: 2 tiles in flight.
    issue_tile(0);
    issue_tile(1);

    #pragma unroll
    for (int t = 0; t < NTILES; ++t) {
        // Tiles 0..t+1 issued so far; in-order completion means asynccnt<=2
        // leaves only tile t+1's two loads possibly outstanding => tile t landed.
        if (t == NTILES - 1) {
            asm volatile("s_wait_asynccnt 0" ::: "memory");
        } else {
            asm volatile("s_wait_asynccnt 2" ::: "memory");
        }
        __syncthreads();   // publish tile t to all waves; proves consume(t-1) done

        // Safe: buf[(t+2)%3] == buf[(t-1)%3], last read during consume(t-1),
        // which every wave completed before this iteration's barrier.
        if (t + 2 < NTILES) issue_tile(t + 2);

        // Consume tile t from LDS: each thread reduces one float4 per half.
        const float* base = &buf[t % NBUF][0];
        float4 a = *(const float4*)(base + tid * 4);              // seg 2t
        float4 b = *(const float4*)(base + TILE / 2 + tid * 4);   // seg 2t+1
        float m0 = fmaxf(fmaxf(a.x, a.y), fmaxf(a.z, a.w));
        float m1 = fmaxf(fmaxf(b.x, b.y), fmaxf(b.z, b.w));
        #pragma unroll
        for (int off = 16; off > 0; off >>= 1) {
            m0 = fmaxf(m0, __shfl_xor(m0, off));
            m1 = fmaxf(m1, __shfl_xor(m1, off));
        }
        if (lane == 0) {
            waveres[(t * 2 + 0) * 8 + wave] = m0;   // wave chunk of segment 2t
            waveres[(t * 2 + 1) * 8 + wave] = m1;   // wave chunk of segment 2t+1
        }
    }
    __syncthreads();

    // Build the 31 outputs from the 16 segment maxes via a pairwise-max tree.
    // Layout: outbuf[0]=p1, [1..2]=p2, [3..6]=p4, [7..14]=p8, [15..30]=p16.
    if (tid < NSEG) {
        float s = waveres[tid * 8];
        #pragma unroll
        for (int k = 1; k < 8; ++k) s = fmaxf(s, waveres[tid * 8 + k]);
        outbuf[15 + tid] = s;
    }
    __syncthreads();
    if (tid < 8) outbuf[7 + tid] = fmaxf(outbuf[15 + 2 * tid], outbuf[16 + 2 * tid]);
    __syncthreads();
    if (tid < 4) outbuf[3 + tid] = fmaxf(outbuf[7 + 2 * tid], outbuf[8 + 2 * tid]);
    __syncthreads();
    if (tid < 2) outbuf[1 + tid] = fmaxf(outbuf[3 + 2 * tid], outbuf[4 + 2 * tid]);
    __syncthreads();
    if (tid == 0) outbuf[0] = fmaxf(outbuf[1], outbuf[2]);
    __syncthreads();

    if (tid < NOUT) out[(size_t)row * NOUT + tid] = outbuf[tid];
}

extern "C" void kernel_launch(void* const* d_in, const int* in_sizes, int n_in,
                              void* d_out, int out_size, void* d_ws, size_t ws_size,
                              hipStream_t stream) {
    const float* x = (const float*)d_in[0];
    float* out = (float*)d_out;
    const int batch = in_sizes[0] / WIDTH;   // 4096
    roi_embed_max_kernel<<<dim3(batch), dim3(NTHREADS), 0, stream>>>(x, out, batch);
}